// My_loss_42279658062027
// MI455X (gfx1250) — compile-verified
//
#include <hip/hip_runtime.h>
#include <math.h>

// CDNA5 / gfx1250, wave32. Batched GEMV total[b,k] = sum_j (H[b,k,j]^2) * (P*y[b,j])
// done with V_WMMA_F32_16X16X4_F32: A = squared-H 16x4 tile (documented layout),
// B = P*y[j0..j0+3] broadcast across all 16 columns -> every column of D holds total[row].

typedef float v2f __attribute__((ext_vector_type(2)));
typedef float v8f __attribute__((ext_vector_type(8)));

#define NB 128         // graphs in batch
#define LL 1024        // links
#define PTX 10.0f
#define PNOISE 6.2946e-14f

__global__ __launch_bounds__(128) void sumrate_wmma_kernel(
    const float* __restrict__ prob,   // [NB*LL, 2]
    const float* __restrict__ H,      // [NB, LL, LL]
    float* __restrict__ part)         // [NB * 64 * 2] tile partial sum-rates
{
    __shared__ float sy[LL];                       // P * y[b, :]

    const int tid  = threadIdx.x;
    const int lane = tid & 31;                     // lane in wave32
    const int wave = tid >> 5;                     // 0..3
    const int b    = blockIdx.x >> 4;              // graph index
    const int tile = ((blockIdx.x & 15) << 2) + wave;  // 0..63 (16-row tile)

    // Stage P*y into LDS (coalesced, strided-2 gather from prob[:,1])
    const float* probB = prob + (size_t)b * (LL * 2);
    for (int i = tid; i < LL; i += 128) sy[i] = PTX * probB[2 * i + 1];
    __syncthreads();

    const float* Hb   = H + (size_t)b * LL * LL;
    const int half    = lane >> 4;                 // 0: K={0,1}, 1: K={2,3}
    const int m       = lane & 15;                 // row within tile
    const int r0      = tile << 4;
    const float* aptr = Hb + (size_t)(r0 + m) * LL + (half << 1);
    const float* bbase = &sy[half << 1];

    v8f acc = {0.f, 0.f, 0.f, 0.f, 0.f, 0.f, 0.f, 0.f};

    for (int j0 = 0; j0 < LL; j0 += 16) {
        // stream-ahead prefetch (~2KB) of this row's H data
        __builtin_prefetch(aptr + j0 + 512, 0, 1);
#pragma unroll
        for (int u = 0; u < 16; u += 4) {
            const int j = j0 + u;
            // A fragment: Hs[r0+m, j + half*2 + {0,1}]  (16x4 f32 A layout)
            v2f av = *(const v2f*)(aptr + j);
            v2f a;
            a.x = av.x * av.x;
            a.y = av.y * av.y;
            // B fragment: P*y[j + half*2 + {0,1}] broadcast across all N columns
            v2f bm;
            bm.x = bbase[j];
            bm.y = bbase[j + 1];
            // D = A*B + C   (8 args: neg_a, A, neg_b, B, c_mod, C, reuse_a, reuse_b)
            acc = __builtin_amdgcn_wmma_f32_16x16x4_f32(
                false, a, false, bm, (short)0, acc, false, false);
        }
    }

    // D layout: VGPR q, lanes 0-15 -> M=q, lanes 16-31 -> M=q+8; value replicated over N.
    // Lanes 0 and 16 each finalize 8 rows of the tile.
    if (m == 0) {
        float s8 = 0.f;
        const int kbase = r0 + (half << 3);
#pragma unroll
        for (int q = 0; q < 8; ++q) {
            const int k  = kbase + q;
            float h      = Hb[(size_t)k * LL + k];     // diagonal (raw, square below)
            float sig    = h * h * sy[k];              // Hs[k,k] * P*y[k]
            float interf = acc[q] - sig;               // total - signal
            s8 += log2f(1.0f + sig / (interf + PNOISE));
        }
        part[((size_t)b * 64 + tile) * 2 + half] = s8;
    }
}

__global__ __launch_bounds__(128) void finalize_kernel(
    const float* __restrict__ part, float* __restrict__ out)
{
    __shared__ float red[128];
    const int t = threadIdx.x;                 // graph index
    const float* p = part + (size_t)t * 128;   // 64 tiles * 2 halves
    float s = 0.f;
#pragma unroll 8
    for (int i = 0; i < 128; ++i) s += p[i];
    red[t] = 1.0f / s;                         // 1 / sum-rate of graph t
    __syncthreads();
    for (int off = 64; off > 0; off >>= 1) {
        if (t < off) red[t] += red[t + off];
        __syncthreads();
    }
    if (t == 0) out[0] = red[0] * (1.0f / (float)NB);
}

extern "C" void kernel_launch(void* const* d_in, const int* in_sizes, int n_in,
                              void* d_out, int out_size, void* d_ws, size_t ws_size,
                              hipStream_t stream)
{
    const float* prob = (const float*)d_in[0];   // [NB*LL, 2] f32
    const float* H    = (const float*)d_in[1];   // [NB, LL, LL] f32
    float* out  = (float*)d_out;                 // scalar f32 loss
    float* part = (float*)d_ws;                  // needs NB*64*2*4 = 64 KB

    // 2048 blocks x 4 waves; each wave = one (graph, 16-row) tile -> 8192 tiles
    sumrate_wmma_kernel<<<dim3(NB * 16), dim3(128), 0, stream>>>(prob, H, part);
    finalize_kernel<<<dim3(1), dim3(128), 0, stream>>>(part, out);
}